// CrossAttention_19387482374277
// MI455X (gfx1250) — compile-verified
//
#include <hip/hip_runtime.h>
#include <hip/hip_bf16.h>

// CrossAttention collapse (B=1):
//   age is broadcast to every pixel  =>  k and v are the same vector for all
//   tokens => each score row is constant => softmax is uniform 1/N =>
//   attended[n] == v_vec = Wv @ age + bv  for every n.
//   output[n, c] = px[n, c] + v_vec[c]     (px = pixel_features^T)
//
// Kernel 1: one wave32 computes v_vec with V_WMMA_F32_16X16X4_F32 (exact f32).
// Kernel 2: memory-bound [128, 16384] -> [16384, 128] transpose + broadcast
//           add, LDS 32x32 tile (stride-33 rows: conflict-free on 64 banks),
//           float4 coalesced global loads and stores.  Traffic: 16 MB total.

typedef __attribute__((ext_vector_type(2))) float v2f;
typedef __attribute__((ext_vector_type(8))) float v8f;

#define C_DIM 128
#define N_PIX 16384   // 16 * 32 * 32
#define AGE_DIM 64

// ---------------------------------------------------------------------------
// v_vec = Wv @ age + bv   via WMMA f32 16x16x4, one wave (32 threads).
//
// For each row-tile t (8 tiles of 16 rows) accumulate over 16 k-steps of 4:
//   A(16x4)[i][k] = Wv[t*16 + i][k0*4 + k]   (A layout: lane = M, lane-half
//                                             selects K pair {0,1}/{2,3})
//   B(4x16)[k][j] = age[k0*4 + k]            (broadcast across columns; B
//                                             layout: VGPR r holds K=r /
//                                             K=r+2 in the two lane halves)
//   C init      = bv[row]  (every column)    => D column 0 is v_vec.
// ---------------------------------------------------------------------------
__global__ __launch_bounds__(32) void ca_vvec_wmma_kernel(
    const float* __restrict__ age,   // [64]
    const float* __restrict__ Wv,    // [128][64] row-major
    const float* __restrict__ bv,    // [128]
    float* __restrict__ vvec)        // [128] out (workspace)
{
  const int l  = threadIdx.x;        // 0..31
  const int m  = l & 15;             // A row / D column index
  const int kb = (l >> 4) * 2;       // K pair base: 0 (lanes 0-15) or 2
  const int mo = (l >> 4) * 8;       // C/D row offset for upper lane half

  for (int t = 0; t < 8; ++t) {
    v8f acc;
#pragma unroll
    for (int r = 0; r < 8; ++r)
      acc[r] = bv[t * 16 + r + mo];  // bias rides the accumulator

#pragma unroll
    for (int k0 = 0; k0 < 16; ++k0) {
      v2f a, b;
      const int row = t * 16 + m;
      a.x = Wv[row * AGE_DIM + k0 * 4 + kb + 0];
      a.y = Wv[row * AGE_DIM + k0 * 4 + kb + 1];
      b.x = age[k0 * 4 + kb + 0];    // same value for every column j
      b.y = age[k0 * 4 + kb + 1];
      acc = __builtin_amdgcn_wmma_f32_16x16x4_f32(
          /*neg_a=*/false, a, /*neg_b=*/false, b,
          /*c_mod=*/(short)0, acc, /*reuse_a=*/false, /*reuse_b=*/false);
    }

    // Extract column N==0: lane 0 holds rows r, lane 16 holds rows r+8.
    if ((l & 15) == 0) {
#pragma unroll
      for (int r = 0; r < 8; ++r)
        vvec[t * 16 + r + mo] = acc[r];
    }
  }
}

// ---------------------------------------------------------------------------
// out[n][c] = px[c][n] + v_vec[c]
// 32(c) x 32(n) tile per block, 256 threads, 4 elements/thread each phase.
// ---------------------------------------------------------------------------
__global__ __launch_bounds__(256) void ca_transpose_add_kernel(
    const float* __restrict__ px,    // [128][16384]
    const float* __restrict__ vvec,  // [128]
    float* __restrict__ out)         // [16384][128]
{
  __shared__ float tile[32][33];     // stride 33: conflict-free both phases

  const int t  = threadIdx.x;
  const int c0 = blockIdx.x * 32;
  const int n0 = blockIdx.y * 32;

  {
    const int cl = t >> 3;           // 0..31  (row of tile = channel)
    const int ng = (t & 7) * 4;      // 0,4,...,28 (pixel within tile)
    const float4 p = *(const float4*)(
        px + (size_t)(c0 + cl) * N_PIX + n0 + ng);
    tile[cl][ng + 0] = p.x;
    tile[cl][ng + 1] = p.y;
    tile[cl][ng + 2] = p.z;
    tile[cl][ng + 3] = p.w;
  }

  __syncthreads();

  {
    const int nl = t >> 3;           // pixel within tile
    const int cg = (t & 7) * 4;      // channel group
    const float4 vb = *(const float4*)(vvec + c0 + cg);
    float4 o;
    o.x = tile[cg + 0][nl] + vb.x;
    o.y = tile[cg + 1][nl] + vb.y;
    o.z = tile[cg + 2][nl] + vb.z;
    o.w = tile[cg + 3][nl] + vb.w;
    *(float4*)(out + (size_t)(n0 + nl) * C_DIM + c0 + cg) = o;
  }
}

extern "C" void kernel_launch(void* const* d_in, const int* in_sizes, int n_in,
                              void* d_out, int out_size, void* d_ws, size_t ws_size,
                              hipStream_t stream) {
  // setup_inputs order:
  // 0: pixel_features [1,128,16,32,32]   (f32)
  // 1: age_features   [1,64]
  // 2: Wq [128,128]  3: bq [128]
  // 4: Wk [128,64]   5: bk [128]
  // 6: Wv [128,64]   7: bv [128]
  const float* px  = (const float*)d_in[0];
  const float* age = (const float*)d_in[1];
  const float* Wv  = (const float*)d_in[6];
  const float* bv  = (const float*)d_in[7];
  float* out  = (float*)d_out;
  float* vvec = (float*)d_ws;   // 128 floats of scratch

  ca_vvec_wmma_kernel<<<1, 32, 0, stream>>>(age, Wv, bv, vvec);

  dim3 grid(C_DIM / 32, N_PIX / 32);   // (4, 512)
  ca_transpose_add_kernel<<<grid, 256, 0, stream>>>(px, vvec, out);
}